// WaveletLayer_5970004542087
// MI455X (gfx1250) — compile-verified
//
#include <hip/hip_runtime.h>

// ---------------------------------------------------------------------------
// WaveletLayer for MI455X (gfx1250)
// Memory-bound streaming transform: 256 MB of HBM traffic @ 23.3 TB/s ~ 11 us.
// Compute is ~0.33 GFLOP total -> no WMMA; the win is the CDNA5 async
// global->LDS copy path + LDS-resident Haar butterflies + NT streaming stores.
// ---------------------------------------------------------------------------

namespace {
constexpr int   kThreads  = 256;   // 8 waves (wave32) per block
constexpr int   kDepth    = 4096;  // row length
constexpr float kInvSqrt2 = 0.70710678118654752440f;
}

typedef int   v4i  __attribute__((vector_size(16)));       // <4 x i32>
typedef float vf2  __attribute__((ext_vector_type(2)));    // native float2

// Async copy 16 bytes per lane from global memory directly into LDS
// (bypasses VGPRs, tracked by ASYNCcnt).
__device__ __forceinline__ void async_copy_b128(const float* gsrc, float* lds_dst) {
#if __has_builtin(__builtin_amdgcn_global_load_async_to_lds_b128)
  __builtin_amdgcn_global_load_async_to_lds_b128(
      (__attribute__((address_space(1))) v4i*)const_cast<float*>(gsrc),
      (__attribute__((address_space(3))) v4i*)lds_dst,
      /*offset=*/0, /*cpol=*/0);
#else
  unsigned lds_off = (unsigned)(unsigned long long)
      (__attribute__((address_space(3))) void*)lds_dst;
  asm volatile("global_load_async_to_lds_b128 %0, %1, off"
               :: "v"(lds_off), "v"(gsrc)
               : "memory");
#endif
}

__device__ __forceinline__ void wait_async_zero() {
#if __has_builtin(__builtin_amdgcn_s_wait_asynccnt)
  __builtin_amdgcn_s_wait_asynccnt(0);
#else
  asm volatile("s_wait_asynccnt 0" ::: "memory");
#endif
}

__global__ __launch_bounds__(kThreads)
void wavelet_row_kernel(const float* __restrict__ x,
                        const float* __restrict__ vb,
                        const float* __restrict__ vg,
                        const float* __restrict__ vs,
                        const int*   __restrict__ perm,
                        float*       __restrict__ out)
{
  // 44 KB LDS / block -> 7 blocks per WGP (320 KB) -> 56 waves resident.
  __shared__ float ls[kDepth];    // staged x row; later the permuted/scaled coeffs d
  __shared__ float c[kDepth];     // analysis coeffs [cA4|cD4|cD3|cD2|cD1]; later r3
  __shared__ float bufA[2048];    // cA ping
  __shared__ float bufB[1024];    // cA pong

  const int    t    = threadIdx.x;
  const size_t base = (size_t)blockIdx.x * kDepth;
  const float* xr   = x   + base;
  float*       orow = out + base;

  // ---- Stage row into LDS via async-tensor path (4 x B128 per lane) ----
#pragma unroll
  for (int k = 0; k < 4; ++k) {
    const int idx = k * (kThreads * 4) + t * 4;   // 16B-aligned, coalesced
    async_copy_b128(xr + idx, &ls[idx]);
  }
  wait_async_zero();
  __syncthreads();

  // ---- Haar analysis L1: (x * vec_b)[4096] -> cA1[2048], cD1 -> c[2048:4096]
#pragma unroll
  for (int j = 0; j < 8; ++j) {
    const int i = t + j * kThreads;
    const float e = ls[2 * i]     * vb[2 * i];
    const float o = ls[2 * i + 1] * vb[2 * i + 1];
    bufA[i]     = (e + o) * kInvSqrt2;
    c[2048 + i] = (e - o) * kInvSqrt2;
  }
  __syncthreads();

  // ---- L2: cA1[2048] -> cA2[1024], cD2 -> c[1024:2048]
#pragma unroll
  for (int j = 0; j < 4; ++j) {
    const int i = t + j * kThreads;
    const float e = bufA[2 * i];
    const float o = bufA[2 * i + 1];
    bufB[i]     = (e + o) * kInvSqrt2;
    c[1024 + i] = (e - o) * kInvSqrt2;
  }
  __syncthreads();

  // ---- L3: cA2[1024] -> cA3[512], cD3 -> c[512:1024]
#pragma unroll
  for (int j = 0; j < 2; ++j) {
    const int i = t + j * kThreads;
    const float e = bufB[2 * i];
    const float o = bufB[2 * i + 1];
    bufA[i]    = (e + o) * kInvSqrt2;
    c[512 + i] = (e - o) * kInvSqrt2;
  }
  __syncthreads();

  // ---- L4: cA3[512] -> cA4 -> c[0:256], cD4 -> c[256:512]
  {
    const int i = t;
    const float e = bufA[2 * i];
    const float o = bufA[2 * i + 1];
    c[i]       = (e + o) * kInvSqrt2;
    c[256 + i] = (e - o) * kInvSqrt2;
  }
  __syncthreads();

  // ---- Permute + gain: d[i] = c[perm[i]] * vec_g[i]  (d lives in ls) ----
#pragma unroll
  for (int j = 0; j < 16; ++j) {
    const int i = t + j * kThreads;
    const int p = perm[i];
    ls[i] = c[p] * vg[i];
  }
  __syncthreads();

  // Reference slice offsets into d:
  //   A0 = d[0:256], D1 = d[2048:2304], D2 = d[1024:1536],
  //   D3 = d[512:1536], D4 = d[256:2304]

  // ---- Synthesis L1: (A0, D1) -> r1[512] in bufB ----
  {
    const int i = t;
    const float a = ls[i];
    const float d = ls[2048 + i];
    bufB[2 * i]     = (a + d) * kInvSqrt2;
    bufB[2 * i + 1] = (a - d) * kInvSqrt2;
  }
  __syncthreads();

  // ---- Synthesis L2: (r1, D2) -> r2[1024] in bufA ----
#pragma unroll
  for (int j = 0; j < 2; ++j) {
    const int i = t + j * kThreads;
    const float a = bufB[i];
    const float d = ls[1024 + i];
    bufA[2 * i]     = (a + d) * kInvSqrt2;
    bufA[2 * i + 1] = (a - d) * kInvSqrt2;
  }
  __syncthreads();

  // ---- Synthesis L3: (r2, D3) -> r3[2048] in c ----
#pragma unroll
  for (int j = 0; j < 4; ++j) {
    const int i = t + j * kThreads;
    const float a = bufA[i];
    const float d = ls[512 + i];
    c[2 * i]     = (a + d) * kInvSqrt2;
    c[2 * i + 1] = (a - d) * kInvSqrt2;
  }
  __syncthreads();

  // ---- Synthesis L4 fused with vec_s scale; NT vf2 streaming stores ----
#pragma unroll
  for (int j = 0; j < 8; ++j) {
    const int i = t + j * kThreads;
    const float a = c[i];
    const float d = ls[256 + i];
    vf2 o;
    o.x = (a + d) * kInvSqrt2 * vs[2 * i];
    o.y = (a - d) * kInvSqrt2 * vs[2 * i + 1];
    __builtin_nontemporal_store(o, (vf2*)(orow + 2 * i));
  }
}

extern "C" void kernel_launch(void* const* d_in, const int* in_sizes, int n_in,
                              void* d_out, int out_size, void* d_ws, size_t ws_size,
                              hipStream_t stream) {
  const float* x    = (const float*)d_in[0];
  const float* vb   = (const float*)d_in[1];
  const float* vg   = (const float*)d_in[2];
  const float* vs   = (const float*)d_in[3];
  const int*   perm = (const int*)d_in[4];
  float*       out  = (float*)d_out;

  const int rows = in_sizes[0] / kDepth;   // BATCH = 8192
  wavelet_row_kernel<<<rows, kThreads, 0, stream>>>(x, vb, vg, vs, perm, out);
}